// MultiSimilarityLoss_25082609008792
// MI455X (gfx1250) — compile-verified
//
#include <hip/hip_runtime.h>
#include <hip/hip_bf16.h>
#include <math.h>

typedef __attribute__((ext_vector_type(2))) float v2f;
typedef __attribute__((ext_vector_type(8))) float v8f;

#define BDIM      4096
#define DDIM      1024
#define PAD       4
#define LDA       (DDIM + PAD)
#define NWAVES    8
#define TPB       (NWAVES * 32)
#define NTILES    (BDIM / 16)

#define THRESH    0.5f
#define MARGIN    0.1f
#define SCALE_POS 2.0f
#define SCALE_NEG 40.0f
#define EPSILON   1e-5f

__global__ __launch_bounds__(TPB)
void ms_loss_main(const float* __restrict__ x,
                  const int*   __restrict__ labels,
                  float*       __restrict__ partial)
{
    __shared__ float As[16 * LDA];               // 16 x 1028 fp32 A-strip (conflict-free reads)
    __shared__ float s_stat0[NWAVES][16];        // per-wave per-row scratch (min / pos_sum)
    __shared__ float s_stat1[NWAVES][16];        // per-wave per-row scratch (max / neg_sum)
    __shared__ float s_minpos[16];
    __shared__ float s_maxneg[16];
    __shared__ int   s_lab[16];
    __shared__ float s_row[16];

    const int tid  = threadIdx.x;
    const int wv   = tid >> 5;
    const int lane = tid & 31;
    const int half = lane >> 4;        // which 16-lane half of the wave
    const int ln   = lane & 15;        // lane within half (= N index / M index)
    const int r0   = blockIdx.x * 16;  // first global row of this strip

    // ---- stage the 16 x 1024 A strip into LDS (coalesced float4) ----
    {
        const float4* xv = (const float4*)(x + (size_t)r0 * DDIM);
        for (int idx = tid; idx < 16 * (DDIM / 4); idx += TPB) {
            int r  = idx >> 8;              // DDIM/4 == 256
            int k4 = idx & 255;
            float4 v = xv[r * (DDIM / 4) + k4];
            float* dst = &As[r * LDA + k4 * 4];
            dst[0] = v.x; dst[1] = v.y; dst[2] = v.z; dst[3] = v.w;
        }
        if (tid < 16) s_lab[tid] = labels[r0 + tid];
    }
    __syncthreads();

    // per-lane row metadata: this lane's C/D fragment covers rows v + 8*half
    int rlab[8], rglob[8];
#pragma unroll
    for (int v = 0; v < 8; ++v) {
        int rr = v + 8 * half;
        rlab[v]  = s_lab[rr];
        rglob[v] = r0 + rr;
    }
    const float* Arow = &As[ln * LDA];
    const int kofs = 2 * half;

    // ================= Pass 1: per-row min_pos / max_neg =================
    float lmin[8], lmax[8];
#pragma unroll
    for (int v = 0; v < 8; ++v) { lmin[v] = __builtin_inff(); lmax[v] = -__builtin_inff(); }

    for (int jt = wv; jt < NTILES; jt += NWAVES) {
        const int cn = jt * 16 + ln;                       // this lane's column
        const float* Bcol = x + (size_t)cn * DDIM;
        const int clab = labels[cn];

        v8f acc = {};
#pragma unroll 4
        for (int kb = 0; kb < DDIM; kb += 4) {
            v2f a; a.x = Arow[kb + kofs]; a.y = Arow[kb + kofs + 1];
            v2f b; b.x = Bcol[kb + kofs]; b.y = Bcol[kb + kofs + 1];
            acc = __builtin_amdgcn_wmma_f32_16x16x4_f32(
                      false, a, false, b, (short)0, acc, false, false);
        }
#pragma unroll
        for (int v = 0; v < 8; ++v) {
            float s   = acc[v];
            bool same = (rlab[v] == clab);
            bool eye  = (rglob[v] == cn);
            bool posv = same && !eye && (s < 1.0f - EPSILON);
            bool neg  = !same;
            if (posv) lmin[v] = fminf(lmin[v], s);
            if (neg)  lmax[v] = fmaxf(lmax[v], s);
        }
    }
    // reduce across the 16 lanes of each half (xor masks < 16 stay within half)
#pragma unroll
    for (int m = 8; m >= 1; m >>= 1) {
#pragma unroll
        for (int v = 0; v < 8; ++v) {
            lmin[v] = fminf(lmin[v], __shfl_xor(lmin[v], m, 32));
            lmax[v] = fmaxf(lmax[v], __shfl_xor(lmax[v], m, 32));
        }
    }
    if (ln == 0) {
#pragma unroll
        for (int v = 0; v < 8; ++v) {
            s_stat0[wv][v + 8 * half] = lmin[v];
            s_stat1[wv][v + 8 * half] = lmax[v];
        }
    }
    __syncthreads();
    if (tid < 16) {
        float mn = __builtin_inff(), mx = -__builtin_inff();
        for (int w = 0; w < NWAVES; ++w) {
            mn = fminf(mn, s_stat0[w][tid]);
            mx = fmaxf(mx, s_stat1[w][tid]);
        }
        s_minpos[tid] = mn;
        s_maxneg[tid] = mx;
    }
    __syncthreads();

    // ================= Pass 2: mined-pair exp sums =================
    float rmn[8], rmx[8], lps[8], lns[8];
#pragma unroll
    for (int v = 0; v < 8; ++v) {
        int rr = v + 8 * half;
        rmn[v] = s_minpos[rr];
        rmx[v] = s_maxneg[rr];
        lps[v] = 0.0f; lns[v] = 0.0f;
    }

    for (int jt = wv; jt < NTILES; jt += NWAVES) {
        const int cn = jt * 16 + ln;
        const float* Bcol = x + (size_t)cn * DDIM;
        const int clab = labels[cn];

        v8f acc = {};
#pragma unroll 4
        for (int kb = 0; kb < DDIM; kb += 4) {
            v2f a; a.x = Arow[kb + kofs]; a.y = Arow[kb + kofs + 1];
            v2f b; b.x = Bcol[kb + kofs]; b.y = Bcol[kb + kofs + 1];
            acc = __builtin_amdgcn_wmma_f32_16x16x4_f32(
                      false, a, false, b, (short)0, acc, false, false);
        }
#pragma unroll
        for (int v = 0; v < 8; ++v) {
            float s   = acc[v];
            bool same = (rlab[v] == clab);
            bool eye  = (rglob[v] == cn);
            bool posv = same && !eye && (s < 1.0f - EPSILON);
            bool neg  = !same;
            bool posk = posv && (s - MARGIN < rmx[v]);   // pos mined vs hardest neg
            bool negk = neg  && (s + MARGIN > rmn[v]);   // neg mined vs hardest pos
            if (posk) lps[v] += expf(-SCALE_POS * (s - THRESH));
            if (negk) lns[v] += expf( SCALE_NEG * (s - THRESH));
        }
    }
#pragma unroll
    for (int m = 8; m >= 1; m >>= 1) {
#pragma unroll
        for (int v = 0; v < 8; ++v) {
            lps[v] += __shfl_xor(lps[v], m, 32);
            lns[v] += __shfl_xor(lns[v], m, 32);
        }
    }
    if (ln == 0) {
#pragma unroll
        for (int v = 0; v < 8; ++v) {
            s_stat0[wv][v + 8 * half] = lps[v];
            s_stat1[wv][v + 8 * half] = lns[v];
        }
    }
    __syncthreads();
    if (tid < 16) {
        float ps = 0.0f, ns = 0.0f;
        for (int w = 0; w < NWAVES; ++w) {
            ps += s_stat0[w][tid];
            ns += s_stat1[w][tid];
        }
        float loss = 0.0f;
        if (ps > 0.0f && ns > 0.0f)   // == has_row (keep-sets non-empty)
            loss = log1pf(ps) / SCALE_POS + log1pf(ns) / SCALE_NEG;
        s_row[tid] = loss;
    }
    __syncthreads();
    if (tid == 0) {
        float t = 0.0f;
#pragma unroll
        for (int i = 0; i < 16; ++i) t += s_row[i];
        partial[blockIdx.x] = t * (1.0f / (float)BDIM);
    }
}

__global__ __launch_bounds__(256)
void ms_loss_reduce(const float* __restrict__ partial, float* __restrict__ out, int n)
{
    __shared__ float s[256];
    int tid = threadIdx.x;
    s[tid] = (tid < n) ? partial[tid] : 0.0f;
    __syncthreads();
    for (int ofs = 128; ofs > 0; ofs >>= 1) {
        if (tid < ofs) s[tid] += s[tid + ofs];
        __syncthreads();
    }
    if (tid == 0) out[0] = s[0];
}

extern "C" void kernel_launch(void* const* d_in, const int* in_sizes, int n_in,
                              void* d_out, int out_size, void* d_ws, size_t ws_size,
                              hipStream_t stream) {
    const float* x      = (const float*)d_in[0];
    const int*   labels = (const int*)d_in[1];
    float*       part   = (float*)d_ws;          // 256 floats of scratch

    ms_loss_main<<<dim3(NTILES), dim3(TPB), 0, stream>>>(x, labels, part);
    ms_loss_reduce<<<dim3(1), dim3(256), 0, stream>>>(part, (float*)d_out, NTILES);
}